// GAT_47906065220065
// MI455X (gfx1250) — compile-verified
//
#include <hip/hip_runtime.h>
#include <cmath>

// ---------------- problem constants (match reference) ----------------
constexpr int N_  = 100000;
constexpr int E_  = 1000000;
constexpr int ET_ = N_ + E_;     // edges + self loops
constexpr int D_  = 64;          // input dim
constexpr int F1_ = 144;         // H1*C1
constexpr int F2_ = 64;          // H2*C2

#define CDIV(a,b) (((a)+(b)-1)/(b))

typedef __attribute__((ext_vector_type(2))) float v2f;
typedef __attribute__((ext_vector_type(8))) float v8f;

// ---------------- helpers ----------------
__device__ __forceinline__ void edge_sd(const int* __restrict__ src,
                                        const int* __restrict__ dst,
                                        int e, int& s, int& d) {
  if (e < E_) { s = src[e]; d = dst[e]; }
  else        { s = e - E_; d = s; }          // self loop
}

// float atomic max via monotonic int/uint mapping (init must be -inf bits)
__device__ __forceinline__ void atomicMaxF(float* addr, float v) {
  if (v >= 0.0f) atomicMax((int*)addr, __float_as_int(v));
  else           atomicMin((unsigned int*)addr, __float_as_uint(v));
}

// ---------------- generic init ----------------
__global__ void k_fill(float* __restrict__ p, int n, float v) {
  int i = blockIdx.x * blockDim.x + threadIdx.x;
  if (i < n) p[i] = v;
}

// ---------------- SSGConv stage ----------------
__global__ void k_deg(const int* __restrict__ dst, float* __restrict__ deg) {
  int e = blockIdx.x * blockDim.x + threadIdx.x;
  if (e < E_) atomicAdd(&deg[dst[e]], 1.0f);
}

__global__ void k_rsqrt_inplace(float* __restrict__ p, int n) {
  int i = blockIdx.x * blockDim.x + threadIdx.x;
  if (i < n) p[i] = rsqrtf(p[i]);
}

__global__ void k_norm(const int* __restrict__ src, const int* __restrict__ dst,
                       const float* __restrict__ dinv, float* __restrict__ norm) {
  int e = blockIdx.x * blockDim.x + threadIdx.x;
  if (e < E_) norm[e] = dinv[src[e]] * dinv[dst[e]];
}

__global__ void k_agg(const int* __restrict__ src, const int* __restrict__ dst,
                      const float* __restrict__ x, const float* __restrict__ norm,
                      const float* __restrict__ dinv, float* __restrict__ agg) {
  int idx = blockIdx.x * blockDim.x + threadIdx.x;
  if (idx >= ET_ * D_) return;
  int e = idx >> 6, c = idx & 63;
  int s, d; float w;
  if (e < E_) { s = src[e]; d = dst[e]; w = norm[e]; }
  else        { s = e - E_; d = s; float di = dinv[s]; w = di * di; }
  atomicAdd(&agg[(size_t)d * D_ + c], x[(size_t)s * D_ + c] * w);
}

__global__ void k_combine(const float* __restrict__ x, float* __restrict__ h) {
  int i = blockIdx.x * blockDim.x + threadIdx.x;
  if (i < N_ * D_) h[i] = 0.5f * x[i] + 0.5f * h[i];   // ALPHA = 0.5, K = 1
}

// ---------------- fp32 WMMA GEMM:  C[M x Nc] = A[M x K] @ W[K x Nc] + bias ----------------
// one wave per 16x16 output tile; V_WMMA_F32_16X16X4_F32 along K.
// A frag (16x4 f32): lanes 0-15 -> M=lane, {K=0,K=1}; lanes 16-31 -> M=lane-16, {K=2,K=3}
// B frag (4x16 f32): lanes 0-15 -> N=lane, {K=0,K=1}; lanes 16-31 -> N=lane-16, {K=2,K=3}
// C/D (16x16 f32, 8 VGPRs): vgpr j -> M=j (lanes 0-15) / M=j+8 (lanes 16-31), N=lane&15
__global__ void k_gemm_bias(const float* __restrict__ A, const float* __restrict__ W,
                            const float* __restrict__ bias, float* __restrict__ C,
                            int K, int Nc) {
  const int wave = blockIdx.x * (blockDim.x >> 5) + (threadIdx.x >> 5);
  const int lane = threadIdx.x & 31;
  const int tilesN = Nc >> 4;
  const int totalTiles = (N_ >> 4) * tilesN;
  if (wave >= totalTiles) return;                 // wave-uniform: EXEC stays full

  const int row0 = (wave / tilesN) << 4;
  const int col0 = (wave % tilesN) << 4;
  const int lm = lane & 15;
  const int lk = (lane >> 4) << 1;                // 0 or 2

  const float* __restrict__ Arow = A + (size_t)(row0 + lm) * K;
  const float* __restrict__ Wcol = W + (col0 + lm);

  v8f acc = {};
  for (int k = 0; k < K; k += 4) {
    v2f a, b;
    a.x = Arow[k + lk];
    a.y = Arow[k + lk + 1];
    b.x = Wcol[(size_t)(k + lk) * Nc];
    b.y = Wcol[(size_t)(k + lk + 1) * Nc];
    acc = __builtin_amdgcn_wmma_f32_16x16x4_f32(false, a, false, b,
                                                (short)0, acc, false, false);
  }

  const int n = col0 + lm;
  const float bn = bias[n];
  const int rbase = row0 + ((lane >> 4) << 3);
#pragma unroll
  for (int j = 0; j < 8; ++j)
    C[(size_t)(rbase + j) * Nc + n] = acc[j] + bn;
}

// ---------------- GATv2 layer 1 (H=12, C=12) ----------------
__global__ void k_logit12(const int* __restrict__ src, const int* __restrict__ dst,
                          const float* __restrict__ xl, const float* __restrict__ xr,
                          const float* __restrict__ att, float* __restrict__ lg,
                          float* __restrict__ m) {
  int idx = blockIdx.x * blockDim.x + threadIdx.x;
  if (idx >= ET_ * 12) return;
  int e = idx / 12, h = idx % 12;
  int s, d; edge_sd(src, dst, e, s, d);
  const float* pl = xl + (size_t)s * F1_ + h * 12;
  const float* pr = xr + (size_t)d * F1_ + h * 12;
  const float* pa = att + h * 12;
  float acc = 0.0f;
#pragma unroll
  for (int c = 0; c < 12; ++c) {
    float t = pl[c] + pr[c];
    t = (t > 0.0f) ? t : 0.2f * t;                // leaky_relu
    acc += t * pa[c];
  }
  lg[idx] = acc;                                  // idx == e*12 + h
  atomicMaxF(&m[d * 12 + h], acc);
}

__global__ void k_expden12(const int* __restrict__ src, const int* __restrict__ dst,
                           const float* __restrict__ m, float* __restrict__ lg,
                           float* __restrict__ den) {
  int idx = blockIdx.x * blockDim.x + threadIdx.x;
  if (idx >= ET_ * 12) return;
  int e = idx / 12, h = idx % 12;
  int s, d; edge_sd(src, dst, e, s, d);
  float a = expf(lg[idx] - m[d * 12 + h]);
  lg[idx] = a;
  atomicAdd(&den[d * 12 + h], a);
}

__global__ void k_accum12(const int* __restrict__ src, const int* __restrict__ dst,
                          const float* __restrict__ lg, const float* __restrict__ xl,
                          float* __restrict__ out) {
  int idx = blockIdx.x * blockDim.x + threadIdx.x;
  if (idx >= ET_ * F1_) return;
  int e = idx / F1_, r = idx % F1_;               // r = h*12 + c
  int h = r / 12;
  int s, d; edge_sd(src, dst, e, s, d);
  atomicAdd(&out[(size_t)d * F1_ + r], lg[e * 12 + h] * xl[(size_t)s * F1_ + r]);
}

__global__ void k_fin12(const float* __restrict__ den, const float* __restrict__ bias,
                        float* __restrict__ out) {
  int idx = blockIdx.x * blockDim.x + threadIdx.x;
  if (idx >= N_ * F1_) return;
  int n = idx / F1_, r = idx % F1_;
  int h = r / 12;
  out[idx] = out[idx] / (den[n * 12 + h] + 1e-16f) + bias[r];
}

// ---------------- GATv2 layer 2 (H=1, C=64): wave-per-edge logit ----------------
__global__ void k_logit1(const int* __restrict__ src, const int* __restrict__ dst,
                         const float* __restrict__ xl, const float* __restrict__ xr,
                         const float* __restrict__ att, float* __restrict__ lg,
                         float* __restrict__ m) {
  int wv = blockIdx.x * (blockDim.x >> 5) + (threadIdx.x >> 5);
  int lane = threadIdx.x & 31;
  if (wv >= ET_) return;                          // wave-uniform exit
  int s, d; edge_sd(src, dst, wv, s, d);
  float acc = 0.0f;
#pragma unroll
  for (int c = lane; c < 64; c += 32) {
    float t = xl[(size_t)s * F2_ + c] + xr[(size_t)d * F2_ + c];
    t = (t > 0.0f) ? t : 0.2f * t;
    acc += t * att[c];
  }
#pragma unroll
  for (int o = 16; o > 0; o >>= 1) acc += __shfl_xor(acc, o, 32);
  if (lane == 0) { lg[wv] = acc; atomicMaxF(&m[d], acc); }
}

__global__ void k_expden1(const int* __restrict__ src, const int* __restrict__ dst,
                          const float* __restrict__ m, float* __restrict__ lg,
                          float* __restrict__ den) {
  int e = blockIdx.x * blockDim.x + threadIdx.x;
  if (e >= ET_) return;
  int s, d; edge_sd(src, dst, e, s, d);
  float a = expf(lg[e] - m[d]);
  lg[e] = a;
  atomicAdd(&den[d], a);
}

__global__ void k_accum1(const int* __restrict__ src, const int* __restrict__ dst,
                         const float* __restrict__ lg, const float* __restrict__ xl,
                         float* __restrict__ num) {
  int idx = blockIdx.x * blockDim.x + threadIdx.x;
  if (idx >= ET_ * F2_) return;
  int e = idx >> 6, c = idx & 63;
  int s, d; edge_sd(src, dst, e, s, d);
  atomicAdd(&num[(size_t)d * F2_ + c], lg[e] * xl[(size_t)s * F2_ + c]);
}

__global__ void k_final(const float* __restrict__ num, const float* __restrict__ den,
                        const float* __restrict__ bias, float* __restrict__ out) {
  int idx = blockIdx.x * blockDim.x + threadIdx.x;
  if (idx >= N_ * F2_) return;
  int n = idx >> 6, c = idx & 63;
  out[idx] = num[idx] / (den[n] + 1e-16f) + bias[c];
}

// ---------------- workspace layout (floats), with lifetime-safe aliasing ----------------
constexpr size_t OFF_DEG  = 0;                         // N   (deg -> dinv; later m2)
constexpr size_t OFF_NORM = OFF_DEG  + N_;             // E   (edge norm; later den2)
constexpr size_t OFF_HBUF = OFF_NORM + E_;             // N*64 (agg -> h; later xl2)
constexpr size_t OFF_X1   = OFF_HBUF + (size_t)N_*D_;  // N*64 (ssg out; later num2)
constexpr size_t OFF_XL1  = OFF_X1   + (size_t)N_*D_;  // N*144 (later xr2 reuses it)
constexpr size_t OFF_XR1  = OFF_XL1  + (size_t)N_*F1_; // N*144 (later out1 reuses it)
constexpr size_t OFF_M1   = OFF_XR1  + (size_t)N_*F1_; // N*12
constexpr size_t OFF_DEN1 = OFF_M1   + (size_t)N_*12;  // N*12
constexpr size_t OFF_LG1  = OFF_DEN1 + (size_t)N_*12;  // (E+N)*12 (later lg2)

extern "C" void kernel_launch(void* const* d_in, const int* in_sizes, int n_in,
                              void* d_out, int out_size, void* d_ws, size_t ws_size,
                              hipStream_t stream) {
  const float* x     = (const float*)d_in[0];
  const int*   ei    = (const int*)  d_in[1];
  const float* Wssg  = (const float*)d_in[2];
  const float* bssg  = (const float*)d_in[3];
  const float* W1l   = (const float*)d_in[4];
  const float* b1l   = (const float*)d_in[5];
  const float* W1r   = (const float*)d_in[6];
  const float* b1r   = (const float*)d_in[7];
  const float* att1  = (const float*)d_in[8];
  const float* bias1 = (const float*)d_in[9];
  const float* W2l   = (const float*)d_in[10];
  const float* b2l   = (const float*)d_in[11];
  const float* W2r   = (const float*)d_in[12];
  const float* b2r   = (const float*)d_in[13];
  const float* att2  = (const float*)d_in[14];
  const float* bias2 = (const float*)d_in[15];

  const int* src = ei;        // edge_index[0]
  const int* dst = ei + E_;   // edge_index[1]

  float* ws   = (float*)d_ws;
  float* deg  = ws + OFF_DEG;    // then dinv (in place)
  float* norm = ws + OFF_NORM;
  float* hbuf = ws + OFF_HBUF;   // agg then h
  float* x1   = ws + OFF_X1;
  float* xl1  = ws + OFF_XL1;
  float* xr1  = ws + OFF_XR1;
  float* m1   = ws + OFF_M1;
  float* den1 = ws + OFF_DEN1;
  float* lg1  = ws + OFF_LG1;
  // aliases (lifetimes verified by launch order below)
  float* out1 = xr1;             // xr1 dead after k_logit12
  float* xl2  = hbuf;            // h dead after x1 GEMM
  float* xr2  = xl1;             // xl1 dead after k_accum12
  float* m2   = deg;             // dinv dead after k_agg
  float* den2 = norm;            // norm dead after k_agg (N <= E)
  float* lg2  = lg1;             // lg1 dead after k_accum12
  float* num2 = x1;              // x1 dead after xl1/xr1 GEMMs
  float* out  = (float*)d_out;

  const int TB = 256;
  const float NEG_INF = -__builtin_huge_valf();

  auto gemm = [&](const float* A, const float* W, const float* b, float* C,
                  int K, int Nc) {
    int tiles = (N_ / 16) * (Nc / 16);
    k_gemm_bias<<<CDIV(tiles, TB / 32), TB, 0, stream>>>(A, W, b, C, K, Nc);
  };

  // ---- SSGConv ----
  k_fill<<<CDIV(N_, TB), TB, 0, stream>>>(deg, N_, 1.0f);            // self loop
  k_fill<<<CDIV(N_*D_, TB), TB, 0, stream>>>(hbuf, N_*D_, 0.0f);
  k_deg<<<CDIV(E_, TB), TB, 0, stream>>>(dst, deg);
  k_rsqrt_inplace<<<CDIV(N_, TB), TB, 0, stream>>>(deg, N_);         // deg -> dinv
  k_norm<<<CDIV(E_, TB), TB, 0, stream>>>(src, dst, deg, norm);
  k_agg<<<CDIV(ET_*D_, TB), TB, 0, stream>>>(src, dst, x, norm, deg, hbuf);
  k_combine<<<CDIV(N_*D_, TB), TB, 0, stream>>>(x, hbuf);            // hbuf = h
  gemm(hbuf, Wssg, bssg, x1, D_, D_);                                // x1 = h@W + b

  // ---- GATv2 layer 1 ----
  gemm(x1, W1l, b1l, xl1, D_, F1_);
  gemm(x1, W1r, b1r, xr1, D_, F1_);
  k_fill<<<CDIV(N_*12, TB), TB, 0, stream>>>(m1, N_*12, NEG_INF);
  k_fill<<<CDIV(N_*12, TB), TB, 0, stream>>>(den1, N_*12, 0.0f);
  k_logit12<<<CDIV(ET_*12, TB), TB, 0, stream>>>(src, dst, xl1, xr1, att1, lg1, m1);
  k_fill<<<CDIV(N_*F1_, TB), TB, 0, stream>>>(out1, N_*F1_, 0.0f);   // after logit12 (aliases xr1)
  k_expden12<<<CDIV(ET_*12, TB), TB, 0, stream>>>(src, dst, m1, lg1, den1);
  k_accum12<<<CDIV(ET_*F1_, TB), TB, 0, stream>>>(src, dst, lg1, xl1, out1);
  k_fin12<<<CDIV(N_*F1_, TB), TB, 0, stream>>>(den1, bias1, out1);

  // ---- GATv2 layer 2 ----
  gemm(out1, W2l, b2l, xl2, F1_, F2_);
  gemm(out1, W2r, b2r, xr2, F1_, F2_);
  k_fill<<<CDIV(N_, TB), TB, 0, stream>>>(m2, N_, NEG_INF);
  k_fill<<<CDIV(N_, TB), TB, 0, stream>>>(den2, N_, 0.0f);
  k_fill<<<CDIV(N_*F2_, TB), TB, 0, stream>>>(num2, N_*F2_, 0.0f);
  k_logit1<<<CDIV(ET_, TB/32), TB, 0, stream>>>(src, dst, xl2, xr2, att2, lg2, m2);
  k_expden1<<<CDIV(ET_, TB), TB, 0, stream>>>(src, dst, m2, lg2, den2);
  k_accum1<<<CDIV(ET_*F2_, TB), TB, 0, stream>>>(src, dst, lg2, xl2, num2);
  k_final<<<CDIV(N_*F2_, TB), TB, 0, stream>>>(num2, den2, bias2, out);
}